// SURFMNet_31310311588236
// MI455X (gfx1250) — compile-verified
//
#include <hip/hip_runtime.h>
#include <hip/hip_bf16.h>

typedef __attribute__((ext_vector_type(2))) float v2f;
typedef __attribute__((ext_vector_type(8))) float v8f;

#define DIM 352      // feature dim D
#define KD  120      // spectral dim K
#define TM 128
#define TN 64
#define TK 16
#define LDA 20       // TK + 4 pad (80B rows: 16B aligned, banks spread)
#define LDBP 132     // paired-B row: 2*TN floats + 4 pad (528B rows: 16B aligned)

// CDNA5 async global->LDS copy (memory -> LDS directly, tracked by ASYNCcnt).
// ldsoff: 32-bit LDS byte offset (low 32 bits of the flat shared address),
// gptr:   64-bit global address (VGPR pair).
#define ASYNC_B128(ldsoff, gptr)                                           \
    asm volatile("global_load_async_to_lds_b128 %0, %1, off"               \
                 :: "v"(ldsoff), "v"(gptr) : "memory")

__device__ __forceinline__ uint32_t lds_addr(const void* p)
{
    // Flat shared-aperture addresses carry the LDS offset in the low 32 bits.
    return (uint32_t)(uintptr_t)p;
}

// ---------------------------------------------------------------------------
// Pack W (352x352 row-major) into pair-interleaved layout:
//   Wp[(k>>1)*704 + 2*n + (k&1)] = W[k*DIM + n]
// so a B fragment {W[k][n], W[k+1][n]} is 8 contiguous bytes -> one
// ds_load_b64 in the GEMM hot loop (no repacking movs).
// ---------------------------------------------------------------------------
__global__ __launch_bounds__(256)
void pack_W(const float* __restrict__ W, float* __restrict__ Wp)
{
    const int idx = blockIdx.x * 256 + threadIdx.x;   // 0 .. DIM*DIM-1
    const int k = idx / DIM;
    const int n = idx - k * DIM;
    Wp[(size_t)(k >> 1) * (2 * DIM) + 2 * n + (k & 1)] = W[idx];
}

// ---------------------------------------------------------------------------
// GEMM: Y = X @ W + bias, X (Mtot x 352) row-major, Wp pair-interleaved.
// Double-buffered LDS pipeline fed by global_load_async_to_lds_b128; fp32
// matrix pipe via v_wmma_f32_16x16x4_f32. Fused epilogue: bias add + per-mtile
// column partial sums / sums-of-squares for BatchNorm.
// 256 threads = 8 waves; wave grid 4(M) x 2(N); each wave owns 32x32 = 2x2
// 16x16 accumulators. Out-of-range rows are clamped at the source (their
// outputs are never stored / never counted); the Wp pad absorbs column
// overreads. All async loads are unpredicated (EXEC all-1s).
// ---------------------------------------------------------------------------
__global__ __launch_bounds__(256)
void gemm_bias_stats(const float* __restrict__ X, const float* __restrict__ Wp,
                     const float* __restrict__ bias, float* __restrict__ Y,
                     float* __restrict__ pSum, float* __restrict__ pSq, int Mtot)
{
    __shared__ float lA[2][TM * LDA];
    __shared__ float lB[2][(TK / 2) * LDBP];
    __shared__ float sSum[TN];
    __shared__ float sSq[TN];

    const int tid   = threadIdx.x;
    const int wave  = tid >> 5;
    const int lane  = tid & 31;
    const int lm    = lane & 15;
    const int half  = lane >> 4;
    const int waveM = wave & 3;
    const int waveN = wave >> 2;
    const int mtile = blockIdx.y;
    const int ntile = blockIdx.x;
    const int mbase = mtile * TM;
    const int nbase = ntile * TN;

    if (tid < TN) { sSum[tid] = 0.0f; sSq[tid] = 0.0f; }

    // ---- per-thread async copy descriptors ----
    // A tile: 128 rows x 16 cols, 8 floats (2 x b128) per thread.
    const int ar   = tid >> 1;
    const int acs  = (tid & 1) * 8;
    int arow = mbase + ar; if (arow > Mtot - 1) arow = Mtot - 1;   // clamp
    const float* aSrc = X + (size_t)arow * DIM + acs;              // +TK per tile
    const uint32_t aD0 = lds_addr(&lA[0][ar * LDA + acs]);
    const uint32_t aD1 = lds_addr(&lA[1][ar * LDA + acs]);

    // B tile: 8 k-pairs x 128 floats (64 cols x 2), 4 floats (1 x b128)/thread.
    const int bkp  = tid >> 5;          // k-pair row 0..7
    const int bseg = lane * 4;          // 0..124
    const float* bSrc = Wp + (size_t)bkp * (2 * DIM) + 2 * nbase + bseg; // +8*704/tile
    const uint32_t bD0 = lds_addr(&lB[0][bkp * LDBP + bseg]);
    const uint32_t bD1 = lds_addr(&lB[1][bkp * LDBP + bseg]);

    v8f zero8 = {0.f,0.f,0.f,0.f,0.f,0.f,0.f,0.f};
    v8f c[2][2];
    c[0][0] = zero8; c[0][1] = zero8; c[1][0] = zero8; c[1][1] = zero8;

    // ---- prologue: prefetch K-tile 0 into buffer 0 ----
    ASYNC_B128(aD0,      aSrc);
    ASYNC_B128(aD0 + 16, aSrc + 4);
    ASYNC_B128(bD0,      bSrc);

    const int NT = DIM / TK;   // 22
    for (int kt = 0; kt < NT; ++kt) {
        const int cur = kt & 1;
        if (kt + 1 < NT) {
            // stream K-tile kt+1 into the other buffer while we compute
            const float* aS = aSrc + (size_t)(kt + 1) * TK;
            const float* bS = bSrc + (size_t)(kt + 1) * (TK / 2) * (2 * DIM);
            const uint32_t aD = cur ? aD0 : aD1;
            const uint32_t bD = cur ? bD0 : bD1;
            ASYNC_B128(aD,      aS);
            ASYNC_B128(aD + 16, aS + 4);
            ASYNC_B128(bD,      bS);
            // wait for this wave's *previous* 3 asyncs (current tile) to land
            asm volatile("s_wait_asynccnt 0x3" ::: "memory");
        } else {
            asm volatile("s_wait_asynccnt 0x0" ::: "memory");
        }
        __syncthreads();   // all waves' current-tile data resident in LDS

        const float* bufA = lA[cur];
        const float* bufB = lB[cur];
        #pragma unroll
        for (int kk = 0; kk < TK; kk += 4) {
            const int ka = kk + 2 * half;      // k index for A pairs
            const int kp = (kk >> 1) + half;   // k-pair row for B
            v2f a[2], b[2];
            #pragma unroll
            for (int i = 0; i < 2; ++i) {
                const float* p = bufA + (waveM * 32 + i * 16 + lm) * LDA + ka;
                a[i].x = p[0];
                a[i].y = p[1];
            }
            #pragma unroll
            for (int j = 0; j < 2; ++j) {
                const float* p = bufB + kp * LDBP + 2 * (waveN * 32 + j * 16 + lm);
                b[j] = *(const v2f*)p;   // {B[ka][n], B[ka+1][n]} contiguous
            }
            #pragma unroll
            for (int i = 0; i < 2; ++i)
                #pragma unroll
                for (int j = 0; j < 2; ++j)
                    c[i][j] = __builtin_amdgcn_wmma_f32_16x16x4_f32(
                        false, a[i], false, b[j], (short)0, c[i][j], false, false);
        }
        __syncthreads();   // everyone done reading `cur` before it is refilled
    }

    // ---- epilogue: bias add, store, per-column partial stats ----
    #pragma unroll
    for (int j = 0; j < 2; ++j) {
        const int colLocal = waveN * 32 + j * 16 + lm;
        const int col = nbase + colLocal;
        const float bj = (col < DIM) ? bias[col] : 0.f;
        float csum = 0.f, csq = 0.f;
        #pragma unroll
        for (int i = 0; i < 2; ++i) {
            #pragma unroll
            for (int r = 0; r < 8; ++r) {
                const int row = mbase + waveM * 32 + i * 16 + half * 8 + r;
                const float v = c[i][j][r] + bj;
                if (row < Mtot && col < DIM) {
                    Y[(size_t)row * DIM + col] = v;
                    csum += v;
                    csq  += v * v;
                }
            }
        }
        csum += __shfl_xor(csum, 16);
        csq  += __shfl_xor(csq, 16);
        if (half == 0 && col < DIM) {
            atomicAdd(&sSum[colLocal], csum);
            atomicAdd(&sSq[colLocal],  csq);
        }
    }
    __syncthreads();
    if (tid < TN) {
        const int col = nbase + tid;
        if (col < DIM) {
            pSum[(size_t)mtile * DIM + col] = sSum[tid];
            pSq[(size_t)mtile * DIM + col]  = sSq[tid];
        }
    }
}

// ---------------------------------------------------------------------------
// Reduce per-mtile partials -> BN affine coefficients a,b per channel.
// ---------------------------------------------------------------------------
__global__ void reduce_stats(const float* __restrict__ pSum, const float* __restrict__ pSq,
                             const float* __restrict__ gamma, const float* __restrict__ beta,
                             float* __restrict__ coefA, float* __restrict__ coefB,
                             int mtiles, float invCnt)
{
    const int c = blockIdx.x * blockDim.x + threadIdx.x;
    if (c >= DIM) return;
    float s = 0.f, q = 0.f;
    for (int m = 0; m < mtiles; ++m) {
        s += pSum[(size_t)m * DIM + c];
        q += pSq[(size_t)m * DIM + c];
    }
    const float mean = s * invCnt;
    const float var  = q * invCnt - mean * mean;   // biased var (ddof=0)
    const float a    = gamma[c] * rsqrtf(var + 1e-3f);
    coefA[c] = a;
    coefB[c] = beta[c] - mean * a;
}

// ---------------------------------------------------------------------------
// Elementwise: T = relu(a*T + b)  (in place)
// ---------------------------------------------------------------------------
__global__ void norm_relu(float* __restrict__ T, const float* __restrict__ cA,
                          const float* __restrict__ cB, size_t n4)
{
    const size_t i = blockIdx.x * (size_t)blockDim.x + threadIdx.x;
    if (i >= n4) return;
    float4 v = ((const float4*)T)[i];
    const int col = (int)((i * 4) % DIM);
    v.x = fmaxf(cA[col + 0] * v.x + cB[col + 0], 0.f);
    v.y = fmaxf(cA[col + 1] * v.y + cB[col + 1], 0.f);
    v.z = fmaxf(cA[col + 2] * v.z + cB[col + 2], 0.f);
    v.w = fmaxf(cA[col + 3] * v.w + cB[col + 3], 0.f);
    ((float4*)T)[i] = v;
}

// ---------------------------------------------------------------------------
// Elementwise: X = relu(a*T + b + X)  (residual, in place on X)
// ---------------------------------------------------------------------------
__global__ void norm_res_relu(float* __restrict__ X, const float* __restrict__ T,
                              const float* __restrict__ cA, const float* __restrict__ cB,
                              size_t n4)
{
    const size_t i = blockIdx.x * (size_t)blockDim.x + threadIdx.x;
    if (i >= n4) return;
    float4 t = ((const float4*)T)[i];
    float4 x = ((const float4*)X)[i];
    const int col = (int)((i * 4) % DIM);
    x.x = fmaxf(cA[col + 0] * t.x + cB[col + 0] + x.x, 0.f);
    x.y = fmaxf(cA[col + 1] * t.y + cB[col + 1] + x.y, 0.f);
    x.z = fmaxf(cA[col + 2] * t.z + cB[col + 2] + x.z, 0.f);
    x.w = fmaxf(cA[col + 3] * t.w + cB[col + 3] + x.w, 0.f);
    ((float4*)X)[i] = x;
}

// ---------------------------------------------------------------------------
// Projection: out[b][d][k] = sum_n feat[b][n][d] * evecs[b][n][k]
// One wave per 16x16 (d,k) tile; WMMA f32 16x16x4 over the N reduction.
// grid: (D/16, ceil(K/16), B), block 32.
// ---------------------------------------------------------------------------
__global__ __launch_bounds__(32)
void proj_kernel(const float* __restrict__ feat, const float* __restrict__ evecs,
                 float* __restrict__ out, int Nv)
{
    const int lane = threadIdx.x;
    const int lm   = lane & 15;
    const int half = lane >> 4;
    const int d0 = blockIdx.x * 16;
    const int k0 = blockIdx.y * 16;
    const int b  = blockIdx.z;
    const float* F = feat  + (size_t)b * Nv * DIM;
    const float* E = evecs + (size_t)b * Nv * KD;
    const int  kc  = k0 + lm;
    const bool kok = kc < KD;

    v8f acc = {0.f,0.f,0.f,0.f,0.f,0.f,0.f,0.f};
    for (int n0 = 0; n0 < Nv; n0 += 4) {
        const int na = n0 + 2 * half;
        v2f a, bb;
        a.x  = (na     < Nv) ? F[(size_t)na       * DIM + d0 + lm] : 0.f;
        a.y  = (na + 1 < Nv) ? F[(size_t)(na + 1) * DIM + d0 + lm] : 0.f;
        bb.x = (kok && na     < Nv) ? E[(size_t)na       * KD + kc] : 0.f;
        bb.y = (kok && na + 1 < Nv) ? E[(size_t)(na + 1) * KD + kc] : 0.f;
        acc = __builtin_amdgcn_wmma_f32_16x16x4_f32(
            false, a, false, bb, (short)0, acc, false, false);
    }
    float* O = out + (size_t)b * DIM * KD;
    if (kok) {
        #pragma unroll
        for (int r = 0; r < 8; ++r)
            O[(size_t)(d0 + half * 8 + r) * KD + kc] = acc[r];
    }
}

// ---------------------------------------------------------------------------
// Gram: pair 0: FtF = Fh^T Fh ; pair 1: FtG = Fh^T Gh ; pair 2: GtG = Gh^T Gh
// (reduce over D=352). grid: (64, B, 3), block 32 (one wave per 16x16 tile).
// ---------------------------------------------------------------------------
__global__ __launch_bounds__(32)
void gram_kernel(const float* __restrict__ Fh, const float* __restrict__ Gh,
                 float* __restrict__ FtF, float* __restrict__ FtG, float* __restrict__ GtG)
{
    const int lane = threadIdx.x;
    const int lm   = lane & 15;
    const int half = lane >> 4;
    const int i    = blockIdx.x >> 3;
    const int j    = blockIdx.x & 7;
    const int b    = blockIdx.y;
    const int pair = blockIdx.z;
    const float* P = (pair == 2) ? Gh : Fh;
    const float* Q = (pair == 0) ? Fh : Gh;
    float* O = (pair == 0) ? FtF : (pair == 1) ? FtG : GtG;
    const float* Pb = P + (size_t)b * DIM * KD;
    const float* Qb = Q + (size_t)b * DIM * KD;
    const int k0 = i * 16, l0 = j * 16;
    const bool kok = (k0 + lm) < KD;
    const bool lok = (l0 + lm) < KD;

    v8f acc = {0.f,0.f,0.f,0.f,0.f,0.f,0.f,0.f};
    for (int d0 = 0; d0 < DIM; d0 += 4) {
        const int da = d0 + 2 * half;
        v2f a, bb;
        a.x  = kok ? Pb[(size_t)da       * KD + k0 + lm] : 0.f;
        a.y  = kok ? Pb[(size_t)(da + 1) * KD + k0 + lm] : 0.f;
        bb.x = lok ? Qb[(size_t)da       * KD + l0 + lm] : 0.f;
        bb.y = lok ? Qb[(size_t)(da + 1) * KD + l0 + lm] : 0.f;
        acc = __builtin_amdgcn_wmma_f32_16x16x4_f32(
            false, a, false, bb, (short)0, acc, false, false);
    }
    float* Ob = O + (size_t)b * KD * KD;
    if (lok) {
        #pragma unroll
        for (int r = 0; r < 8; ++r) {
            const int krow = k0 + half * 8 + r;
            if (krow < KD) Ob[(size_t)krow * KD + (l0 + lm)] = acc[r];
        }
    }
}

// ---------------------------------------------------------------------------
// Solve: one WG per (batch, system). In-place Gauss-Jordan inversion of the
// SPD 120x120 Gram matrix in LDS (57.6 KB), then:
//   which==0: C1[b][i][j] = sum_t inv(FtF)[j][t] * FtG[b][t][i]
//   which==1: C2[b][i][j] = sum_t inv(GtG)[j][t] * FtG[b][i][t]  (GtF = FtG^T)
// ---------------------------------------------------------------------------
__global__ __launch_bounds__(256)
void solve_kernel(const float* __restrict__ FtF, const float* __restrict__ FtG,
                  const float* __restrict__ GtG,
                  float* __restrict__ C1, float* __restrict__ C2)
{
    __shared__ float As[KD * KD];
    __shared__ float fbuf[KD];
    __shared__ float dsh;

    const int b     = blockIdx.x >> 1;
    const int which = blockIdx.x & 1;
    const int tid   = threadIdx.x;
    const float* Asrc = (which ? GtG : FtF) + (size_t)b * KD * KD;

    for (int idx = tid; idx < KD * KD; idx += 256) As[idx] = Asrc[idx];
    __syncthreads();

    for (int p = 0; p < KD; ++p) {
        if (tid == 0) dsh = 1.0f / As[p * KD + p];
        __syncthreads();
        const float d = dsh;
        for (int i2 = tid; i2 < KD; i2 += 256) fbuf[i2] = As[i2 * KD + p];
        __syncthreads();
        for (int jj = tid; jj < KD; jj += 256)
            if (jj != p) As[p * KD + jj] *= d;
        __syncthreads();
        for (int idx = tid; idx < KD * KD; idx += 256) {
            const int i2 = idx / KD;
            const int jj = idx - i2 * KD;
            if (i2 == p) continue;
            if (jj == p) As[idx] = -fbuf[i2] * d;
            else         As[idx] -= fbuf[i2] * As[p * KD + jj];
        }
        __syncthreads();
        if (tid == 0) As[p * KD + p] = d;
        __syncthreads();
    }

    const float* Bsrc = FtG + (size_t)b * KD * KD;
    float* Cout = (which ? C2 : C1) + (size_t)b * KD * KD;
    for (int idx = tid; idx < KD * KD; idx += 256) {
        const int i2 = idx / KD;
        const int jj = idx - i2 * KD;
        float s = 0.f;
        if (which == 0) {
            for (int t = 0; t < KD; ++t) s += As[jj * KD + t] * Bsrc[(size_t)t * KD + i2];
        } else {
            for (int t = 0; t < KD; ++t) s += As[jj * KD + t] * Bsrc[(size_t)i2 * KD + t];
        }
        Cout[idx] = s;
    }
}

// ---------------------------------------------------------------------------
extern "C" void kernel_launch(void* const* d_in, const int* in_sizes, int n_in,
                              void* d_out, int out_size, void* d_ws, size_t ws_size,
                              hipStream_t stream)
{
    (void)in_sizes; (void)n_in; (void)out_size; (void)ws_size;

    const float* feat_x  = (const float*)d_in[0];
    const float* feat_y  = (const float*)d_in[1];
    const float* evecs_x = (const float*)d_in[2];
    const float* evecs_y = (const float*)d_in[3];
    const float* W1      = (const float*)d_in[4];
    const float* b1      = (const float*)d_in[5];
    const float* W2      = (const float*)d_in[6];
    const float* b2      = (const float*)d_in[7];
    const float* gamma   = (const float*)d_in[8];
    const float* beta    = (const float*)d_in[9];

    float* out = (float*)d_out;
    float* ws  = (float*)d_ws;

    const int B = 8, Nv = 6890;
    const size_t featElems = (size_t)B * Nv * DIM;   // 19,402,240
    const int Mtot   = B * Nv;                       // 55120
    const int mtiles = (Mtot + TM - 1) / TM;         // 431
    const int ntiles = (DIM + TN - 1) / TN;          // 6

    // d_out layout: C1 | C2 | fx | fy  (reference return order)
    float* C1 = out;
    float* C2 = out + (size_t)B * KD * KD;
    float* fx = out + 2 * (size_t)B * KD * KD;
    float* fy = fx + featElems;

    // workspace layout
    float* T1   = ws;                                  // GEMM1 output
    float* T2   = ws + featElems;                      // GEMM2 output
    float* pSum = T2 + featElems;
    float* pSq  = pSum + (size_t)mtiles * DIM;
    float* cA   = pSq  + (size_t)mtiles * DIM;
    float* cB   = cA + DIM;
    float* Wp   = cB + DIM;                            // packed weights +64 pad
    // funcmap buffers reuse T1's region (refine is done by then)
    float* Fh  = T1;
    float* Gh  = Fh  + (size_t)B * DIM * KD;
    float* FtF = Gh  + (size_t)B * DIM * KD;
    float* FtG = FtF + (size_t)B * KD * KD;
    float* GtG = FtG + (size_t)B * KD * KD;

    // refine happens in place in the fx/fy output regions
    hipMemcpyAsync(fx, feat_x, featElems * sizeof(float), hipMemcpyDeviceToDevice, stream);
    hipMemcpyAsync(fy, feat_y, featElems * sizeof(float), hipMemcpyDeviceToDevice, stream);

    const dim3 gGemm(ntiles, mtiles);
    const int packBlocks = (DIM * DIM) / 256;          // 484
    const float invCnt = 1.0f / (float)Mtot;
    const size_t n4 = featElems / 4;
    const int ewBlocks = (int)((n4 + 255) / 256);

    for (int f = 0; f < 2; ++f) {
        float* X = f ? fy : fx;
        for (int blk = 0; blk < 7; ++blk) {
            const float* W1b = W1 + (size_t)blk * DIM * DIM;
            const float* b1b = b1 + (size_t)blk * DIM;
            const float* W2b = W2 + (size_t)blk * DIM * DIM;
            const float* b2b = b2 + (size_t)blk * DIM;
            const float* gm  = gamma + (size_t)blk * DIM;
            const float* bt  = beta  + (size_t)blk * DIM;

            pack_W<<<packBlocks, 256, 0, stream>>>(W1b, Wp);
            gemm_bias_stats<<<gGemm, 256, 0, stream>>>(X, Wp, b1b, T1, pSum, pSq, Mtot);
            reduce_stats<<<dim3(3), 128, 0, stream>>>(pSum, pSq, gm, bt, cA, cB, mtiles, invCnt);
            norm_relu<<<ewBlocks, 256, 0, stream>>>(T1, cA, cB, n4);

            pack_W<<<packBlocks, 256, 0, stream>>>(W2b, Wp);
            gemm_bias_stats<<<gGemm, 256, 0, stream>>>(T1, Wp, b2b, T2, pSum, pSq, Mtot);
            reduce_stats<<<dim3(3), 128, 0, stream>>>(pSum, pSq, gm, bt, cA, cB, mtiles, invCnt);
            norm_res_relu<<<ewBlocks, 256, 0, stream>>>(X, T2, cA, cB, n4);
        }
    }

    // funcmap
    const dim3 gProj(DIM / 16, (KD + 15) / 16, B);
    proj_kernel<<<gProj, 32, 0, stream>>>(fx, evecs_x, Fh, Nv);
    proj_kernel<<<gProj, 32, 0, stream>>>(fy, evecs_y, Gh, Nv);
    const dim3 gGram(64, B, 3);
    gram_kernel<<<gGram, 32, 0, stream>>>(Fh, Gh, FtF, FtG, GtG);
    solve_kernel<<<2 * B, 256, 0, stream>>>(FtF, FtG, GtG, C1, C2);
}